// SerializedPooling_62294205661682
// MI455X (gfx1250) — compile-verified
//
#include <hip/hip_runtime.h>
#include <hip/hip_bf16.h>
#include <math.h>

typedef __attribute__((ext_vector_type(2))) float v2f;
typedef __attribute__((ext_vector_type(8))) float v8f;
typedef __attribute__((ext_vector_type(3))) float v3f;
typedef __attribute__((ext_vector_type(4))) int   v4i;
typedef __attribute__((ext_vector_type(4))) unsigned v4u;

#define C_IN 64
#define C_OUT 128
#define OUT_STRIDE 131      // 128 features + 3 coords per pooled row
#define BN_EPS 1e-3f
#define WL_STRIDE 66        // LDS pad: even (8B-aligned b64) + bank-spread

// Branchless monotone float<->uint encoding so segment-max is a single
// hardware GLOBAL_ATOMIC_MAX_U32 (no CAS loop).
__device__ __forceinline__ unsigned fenc(float f) {
    unsigned u = __float_as_uint(f);
    return u ^ ((unsigned)((int)u >> 31) | 0x80000000u);
}
__device__ __forceinline__ float fdec(unsigned k) {
    unsigned u = k ^ (~(unsigned)((int)k >> 31) | 0x80000000u);
    return __uint_as_float(u);
}

// Vectorized zero: n dwords = n4 * 4 + tail (thread 0 handles the tail).
__global__ void zero_kernel(unsigned* __restrict__ p, long long n) {
    long long i = (long long)blockIdx.x * blockDim.x + threadIdx.x;
    long long n4 = n >> 2;
    if (i < n4) ((v4u*)p)[i] = (v4u){0u, 0u, 0u, 0u};
    if (i == 0)
        for (long long t = n4 << 2; t < n; ++t) p[t] = 0u;
}

// Fused GEMM (64 feat rows x W[64x128]) + bias + encoded atomic segment-max
// into d_out columns 0..127.  One wave = four 16x16 f32 WMMA tiles; K=64 as
// 16 x V_WMMA_F32_16X16X4_F32 split over two independent accumulator chains.
// W is transposed into LDS once per block so B operands are ds_load_2addr_b64.
// FULL=true: branch-free fast path over complete 64-row blocks.
// FULL=false: single guarded tail block starting at rowOrigin.
template <bool FULL>
__global__ __launch_bounds__(256) void gemm_pool_kernel(
        const float* __restrict__ feat, const float* __restrict__ W,
        const float* __restrict__ bias, const int* __restrict__ code0,
        unsigned* __restrict__ out, long long rowOrigin, long long N) {
    __shared__ float Wl[C_OUT * WL_STRIDE];          // ~33 KB of 320 KB/WGP

    const int tid   = threadIdx.x;
    const int lane  = tid & 31;
    const int w     = tid >> 5;      // wave 0..7
    const int wr    = w >> 1;        // row tile 0..3
    const int wc    = w & 1;         // col sub-tile 0..1
    const int mloc  = lane & 15;
    const int khalf = lane >> 4;     // lanes 16..31: K+2/K+3 (A,B), M+8 (D)

    // Stage W transposed: Wl[c*WL_STRIDE + k] = W[k*128 + c]
    for (int i = tid; i < C_IN * C_OUT; i += 256) {
        int k = i >> 7, c = i & 127;
        Wl[c * WL_STRIDE + k] = W[i];
    }
    __syncthreads();

    const long long rowbase = rowOrigin + (long long)blockIdx.x * 64 + wr * 16;

    // A-matrix: lane holds row mloc; K = 4*kk + 2*khalf + {0,1}
    long long arow = rowbase + mloc;
    if (!FULL && arow >= N) arow = N - 1;
    const float* frow = feat + arow * C_IN;
    v2f a[16];
#pragma unroll
    for (int kk = 0; kk < 16; ++kk)
        a[kk] = *(const v2f*)(frow + kk * 4 + khalf * 2);

    // Cluster ids for the 8 output rows this lane owns (D: m = v + 8*khalf).
    // Rows are consecutive & 32B-aligned -> two b128 loads in the fast path.
    int cl[8];
    if (FULL) {
        const int* cp = code0 + rowbase + khalf * 8;
        *(v4i*)&cl[0] = *(const v4i*)(cp);
        *(v4i*)&cl[4] = *(const v4i*)(cp + 4);
    } else {
#pragma unroll
        for (int v = 0; v < 8; ++v) {
            long long r = rowbase + v + khalf * 8;
            cl[v] = code0[r < N ? r : (N - 1)];
        }
    }

#pragma unroll
    for (int cj = 0; cj < 4; ++cj) {
        const int col = (cj * 2 + wc) * 16 + mloc;   // this lane's column
        const float* wcol = &Wl[col * WL_STRIDE];
        v8f acc0 = {}, acc1 = {};
#pragma unroll
        for (int kk = 0; kk < 16; kk += 2) {         // two independent chains
            v2f b0 = *(const v2f*)(wcol + kk * 4 + khalf * 2);
            v2f b1 = *(const v2f*)(wcol + (kk + 1) * 4 + khalf * 2);
            acc0 = __builtin_amdgcn_wmma_f32_16x16x4_f32(
                       false, a[kk],     false, b0, (short)0, acc0, false, false);
            acc1 = __builtin_amdgcn_wmma_f32_16x16x4_f32(
                       false, a[kk + 1], false, b1, (short)0, acc1, false, false);
        }
        const float bv = bias[col];
        v8f s = acc0 + acc1;
#pragma unroll
        for (int v = 0; v < 8; ++v) {
            if (FULL || (rowbase + v + khalf * 8 < N))
                atomicMax(out + (size_t)cl[v] * OUT_STRIDE + col,
                          fenc(s[v] + bv));
        }
    }
}

// Coord segment-sum (GLOBAL_ATOMIC_ADD_F32) into d_out cols 128..130 plus
// per-cluster counts in workspace.  Single 12-byte coord load per point.
__global__ void coord_count_kernel(const float* __restrict__ coord,
                                   const int* __restrict__ code0,
                                   float* __restrict__ out,
                                   unsigned* __restrict__ counts, int n) {
    int i = blockIdx.x * blockDim.x + threadIdx.x;
    if (i >= n) return;
    int c = code0[i];
    atomicAdd(&counts[c], 1u);
    v3f p = *(const v3f*)(coord + (size_t)i * 3);
    float* dst = out + (size_t)c * OUT_STRIDE + C_OUT;
    atomicAdd(&dst[0], p.x);
    atomicAdd(&dst[1], p.y);
    atomicAdd(&dst[2], p.z);
}

// Per-channel sum / sum-of-squares over the M pooled rows (decoded keys).
// threadIdx.x == channel -> 128 consecutive dwords per row, fully coalesced.
__global__ __launch_bounds__(128) void stats_kernel(
        const unsigned* __restrict__ out, float* __restrict__ ssum,
        float* __restrict__ ssq, int M, int rows_per_block) {
    const int c = threadIdx.x;
    long long r0 = (long long)blockIdx.x * rows_per_block;
    float s = 0.f, q = 0.f;
    for (int r = 0; r < rows_per_block; ++r) {
        long long row = r0 + r;
        if (row >= M) break;
        float x = fdec(out[row * OUT_STRIDE + c]);
        s += x; q += x * x;
    }
    atomicAdd(&ssum[c], s);
    atomicAdd(&ssq[c], q);
}

// In-place: decode keys, BatchNorm (biased batch stats) + exact-erf GELU on
// cols 0..127; divide coord sums by counts on cols 128..130.
__global__ void finalize_kernel(float* __restrict__ out,
                                const float* __restrict__ ssum,
                                const float* __restrict__ ssq,
                                const unsigned* __restrict__ counts,
                                const float* __restrict__ gamma,
                                const float* __restrict__ beta, int M) {
    long long idx = (long long)blockIdx.x * blockDim.x + threadIdx.x;
    long long total = (long long)M * OUT_STRIDE;
    if (idx >= total) return;
    int c = (int)(idx % OUT_STRIDE);
    long long m = idx / OUT_STRIDE;
    if (c < C_OUT) {
        float invM  = 1.0f / (float)M;
        float mean  = ssum[c] * invM;
        float var   = ssq[c] * invM - mean * mean;
        float x     = fdec(__float_as_uint(out[idx]));
        x = (x - mean) * rsqrtf(var + BN_EPS) * gamma[c] + beta[c];
        out[idx] = 0.5f * x * (1.0f + erff(x * 0.70710678118654752f));
    } else {
        float cnt = (float)counts[m];
        out[idx] = out[idx] / cnt;
    }
}

extern "C" void kernel_launch(void* const* d_in, const int* in_sizes, int n_in,
                              void* d_out, int out_size, void* d_ws, size_t ws_size,
                              hipStream_t stream) {
    const float* feat  = (const float*)d_in[0];
    const float* coord = (const float*)d_in[1];
    const int*   code0 = (const int*)d_in[2];
    const float* W     = (const float*)d_in[3];
    const float* b     = (const float*)d_in[4];
    const float* gamma = (const float*)d_in[5];
    const float* beta  = (const float*)d_in[6];

    const long long N = (long long)in_sizes[0] / C_IN;   // 2,000,000
    const int M = out_size / OUT_STRIDE;                 // 1,000,000

    unsigned* counts = (unsigned*)d_ws;                        // M uints
    float*    ssum   = (float*)((char*)d_ws + (size_t)M * 4);  // 128 floats
    float*    ssq    = ssum + C_OUT;                           // 128 floats

    // 1) zero output (keys + coord sums) and workspace accumulators
    long long zout = (long long)M * OUT_STRIDE;
    zero_kernel<<<(unsigned)((zout / 4 + 255) / 256), 256, 0, stream>>>(
        (unsigned*)d_out, zout);
    long long zws = (long long)M + 2 * C_OUT;
    zero_kernel<<<(unsigned)((zws / 4 + 255) / 256), 256, 0, stream>>>(
        counts, zws);

    // 2) fused WMMA GEMM + bias + atomic segment-max
    long long fullBlocks = N / 64;                 // branch-free fast path
    if (fullBlocks > 0)
        gemm_pool_kernel<true><<<(unsigned)fullBlocks, 256, 0, stream>>>(
            feat, W, b, code0, (unsigned*)d_out, 0, N);
    if (N - fullBlocks * 64 > 0)                   // guarded tail block
        gemm_pool_kernel<false><<<1, 256, 0, stream>>>(
            feat, W, b, code0, (unsigned*)d_out, fullBlocks * 64, N);

    // 3) coord segment-sum + counts
    coord_count_kernel<<<(unsigned)((N + 255) / 256), 256, 0, stream>>>(
        coord, code0, (float*)d_out, counts, (int)N);

    // 4) per-channel batch stats over pooled rows
    const int rpb = 500;
    stats_kernel<<<(unsigned)((M + rpb - 1) / rpb), 128, 0, stream>>>(
        (const unsigned*)d_out, ssum, ssq, M, rpb);

    // 5) BN + GELU + coord mean, in place
    finalize_kernel<<<(unsigned)((zout + 255) / 256), 256, 0, stream>>>(
        (float*)d_out, ssum, ssq, counts, gamma, beta, M);
}